// Embedding_SPD_36661840838909
// MI455X (gfx1250) — compile-verified
//
#include <hip/hip_runtime.h>
#include <hip/hip_bf16.h>
#include <math.h>

// ---------------------------------------------------------------------------
// SPD embedding pipeline for gfx1250 (MI455X), built entirely on
// v_wmma_f32_16x16x4_f32.  All eigendecompositions of the reference are
// replaced by matmul-only iterations (Newton-Schulz sqrt, ISS logm,
// scaling-and-squaring expm), so the whole pipeline maps onto WMMA.
// Sample tiles are brought into LDS with GLOBAL_LOAD_ASYNC_TO_LDS_B128
// (ASYNCcnt) to avoid the VGPR round-trip on the 2 x 102400 hot waves.
// ---------------------------------------------------------------------------

typedef float v2f __attribute__((ext_vector_type(2)));
typedef float v8f __attribute__((ext_vector_type(8)));

#define N_B     32
#define S_TOT   3200
#define D0      136
#define D_FULL  152
#define DP      160          // padded leading dim (10 x 16 tiles)
#define SQRT2   1.41421356237309515f

#define NS_ITERS  8          // Newton-Schulz iterations for sqrtm/invsqrtm
#define LOG_ROOTS 3          // inverse scaling-and-squaring square roots
#define LOG_TERMS 7          // Mercator terms for log(I+E)
#define EXP_SQ    4          // expm squarings (scale by 1/16)
#define EXP_TERMS 8          // expm Taylor terms

// workspace layout (float offsets)
#define OFF_RM      0u
#define OFF_SQ1     8192u
#define OFF_ISQ1    16384u
#define OFF_GT      24576u
#define OFF_XMEAN   32768u
#define OFF_SQ2     40960u
#define OFF_ISQ2    49152u
#define OFF_LTMEAN  57344u        // 32*136
#define OFF_WSQ     61952u        // 256
#define OFF_RBIG    65536u        // 160*160
#define OFF_SCR0    91136u
#define OFF_SCR1    116736u
#define OFF_SCR2    142336u
#define OFF_COV     167936u       // 32*160*160
#define OFF_E       987136u       // 32*160*160
#define OFF_XC      1806336u      // 32*3200*160

// ---------------------------------------------------------------------------
// Async copy of one 1KB (256-float) matrix from global memory into LDS.
// 32 lanes x 16B x 2 instructions; INST_OFFSET applies to both the global
// and the LDS address (ISA 15.18), so a single vaddr/vdst pair suffices.
// Completion is tracked by ASYNCcnt; call wait_async0() before use.
// ---------------------------------------------------------------------------
__device__ __forceinline__ void async_load_1k(const float* gsrc, float* ldst,
                                              int lane) {
  unsigned lds = (unsigned)(size_t)ldst + (unsigned)lane * 16u;
  unsigned long long ga =
      (unsigned long long)(size_t)gsrc + (unsigned long long)lane * 16ull;
  asm volatile(
      "global_load_async_to_lds_b128 %0, %1, off\n\t"
      "global_load_async_to_lds_b128 %0, %1, off offset:512"
      :
      : "v"(lds), "v"(ga)
      : "memory");
}

__device__ __forceinline__ void wait_async0() {
  asm volatile("s_wait_asynccnt 0" ::: "memory");
}

// ---------------------------------------------------------------------------
// One-wave 16x16 f32 matmul: D = A @ B, matrices row-major in LDS.
// A-frag (16x4): lane l -> m=l&15, h=l>>4, vgpr0=A[m][k+2h], vgpr1=A[m][k+2h+1]
// B-frag (4x16): lane l -> n=l&15, vgpr0=B[k+2h][n], vgpr1=B[k+2h+1][n]
// C/D        : vgpr r -> row r+8h, col l&15
// In-place D==A or D==B is safe: all LDS loads are issued before the stores
// and same-wave LDS ops complete in order.
// ---------------------------------------------------------------------------
__device__ __forceinline__ void mm16(const float* A, const float* B, float* D,
                                     int lane) {
  const int m = lane & 15, h = lane >> 4;
  v8f acc = {};
#pragma unroll
  for (int k = 0; k < 16; k += 4) {
    v2f a, b;
    a.x = A[m * 16 + k + 2 * h];
    a.y = A[m * 16 + k + 2 * h + 1];
    b.x = B[(k + 2 * h) * 16 + m];
    b.y = B[(k + 2 * h + 1) * 16 + m];
    acc = __builtin_amdgcn_wmma_f32_16x16x4_f32(false, a, false, b, (short)0,
                                                acc, false, false);
  }
#pragma unroll
  for (int r = 0; r < 8; ++r) D[(r + 8 * h) * 16 + m] = acc[r];
}

// Y = sqrtm(M), Z = invsqrtm(M) via Newton-Schulz (trace-normalized).
__device__ void spd_sqrt_invsqrt(const float* M, float* Y, float* Z, float* T,
                                 int lane) {
  float c = 0.f;
  for (int i = 0; i < 16; ++i) c += M[i * 17];
  c = fmaxf(c, 1e-8f);
  const float inv_c = 1.f / c;
  for (int i = 0; i < 8; ++i) {
    int e = lane * 8 + i, r = e >> 4, cc = e & 15;
    Y[e] = M[e] * inv_c;
    Z[e] = (r == cc) ? 1.f : 0.f;
  }
  for (int it = 0; it < NS_ITERS; ++it) {
    mm16(Z, Y, T, lane);
    for (int i = 0; i < 8; ++i) {
      int e = lane * 8 + i, r = e >> 4, cc = e & 15;
      T[e] = ((r == cc) ? 3.f : 0.f) - T[e];
    }
    mm16(Y, T, Y, lane);
    mm16(T, Z, Z, lane);
    for (int i = 0; i < 8; ++i) {
      int e = lane * 8 + i;
      Y[e] *= 0.5f;
      Z[e] *= 0.5f;
    }
  }
  const float sc = sqrtf(c), isc = 1.f / sc;
  for (int i = 0; i < 8; ++i) {
    int e = lane * 8 + i;
    Y[e] *= sc;
    Z[e] *= isc;
  }
}

// OUT = logm(A) (SPD), inverse scaling-and-squaring + Mercator. A destroyed.
__device__ void spd_logm(float* A, float* OUT, float* t1, float* t2, float* t3,
                         int lane) {
  for (int rt = 0; rt < LOG_ROOTS; ++rt) {
    float c = 0.f;
    for (int i = 0; i < 16; ++i) c += A[i * 17];
    c = fmaxf(c, 1e-8f);
    const float inv_c = 1.f / c;
    for (int i = 0; i < 8; ++i) {
      int e = lane * 8 + i, r = e >> 4, cc = e & 15;
      t1[e] = A[e] * inv_c;
      t2[e] = (r == cc) ? 1.f : 0.f;
    }
    for (int it = 0; it < NS_ITERS; ++it) {
      mm16(t2, t1, t3, lane);
      for (int i = 0; i < 8; ++i) {
        int e = lane * 8 + i, r = e >> 4, cc = e & 15;
        t3[e] = ((r == cc) ? 3.f : 0.f) - t3[e];
      }
      mm16(t1, t3, t1, lane);
      mm16(t3, t2, t2, lane);
      for (int i = 0; i < 8; ++i) {
        int e = lane * 8 + i;
        t1[e] *= 0.5f;
        t2[e] *= 0.5f;
      }
    }
    const float sc = sqrtf(c);
    for (int i = 0; i < 8; ++i) {
      int e = lane * 8 + i;
      A[e] = t1[e] * sc;
    }
  }
  // E = A - I  (kept in A)
  for (int i = 0; i < 8; ++i) {
    int e = lane * 8 + i, r = e >> 4, cc = e & 15;
    A[e] -= (r == cc) ? 1.f : 0.f;
  }
  // Horner for log(I+E) = E @ H1,  H_k = (1/k)I - E@H_{k+1}
  for (int i = 0; i < 8; ++i) {
    int e = lane * 8 + i, r = e >> 4, cc = e & 15;
    t1[e] = (r == cc) ? (1.f / (float)LOG_TERMS) : 0.f;
  }
  for (int k = LOG_TERMS - 1; k >= 1; --k) {
    mm16(A, t1, t2, lane);
    const float ik = 1.f / (float)k;
    for (int i = 0; i < 8; ++i) {
      int e = lane * 8 + i, r = e >> 4, cc = e & 15;
      t1[e] = ((r == cc) ? ik : 0.f) - t2[e];
    }
  }
  mm16(A, t1, OUT, lane);
  const float sc = (float)(1 << LOG_ROOTS);
  for (int i = 0; i < 8; ++i) {
    int e = lane * 8 + i;
    OUT[e] *= sc;
  }
}

// OUT = expm(pre*M), scaling-and-squaring + Taylor. A,P scratch (!=M,OUT).
__device__ void spd_expm(const float* M, float* OUT, float* A, float* P,
                         int lane, float pre) {
  const float s = pre * (1.f / (float)(1 << EXP_SQ));
  for (int i = 0; i < 8; ++i) {
    int e = lane * 8 + i, r = e >> 4, cc = e & 15;
    float a = M[e] * s;
    A[e] = a;
    P[e] = a;
    OUT[e] = ((r == cc) ? 1.f : 0.f) + a;
  }
  for (int k = 2; k <= EXP_TERMS; ++k) {
    mm16(P, A, P, lane);
    const float ik = 1.f / (float)k;
    for (int i = 0; i < 8; ++i) {
      int e = lane * 8 + i;
      P[e] *= ik;
      OUT[e] += P[e];
    }
  }
  for (int sq = 0; sq < EXP_SQ; ++sq) mm16(OUT, OUT, OUT, lane);
}

__device__ __forceinline__ void tri_rc(int idx, int* r, int* c) {
  int rr = (int)((sqrtf(8.f * (float)idx + 1.f) - 1.f) * 0.5f);
  while ((rr + 1) * (rr + 2) / 2 <= idx) ++rr;
  while (rr * (rr + 1) / 2 > idx) --rr;
  *r = rr;
  *c = idx - rr * (rr + 1) / 2;
}

// ---------------------------------------------------------------------------
// kernels
// ---------------------------------------------------------------------------
__global__ void k_zero(float* __restrict__ p, int n) {
  int i = blockIdx.x * blockDim.x + threadIdx.x;
  if (i < n) p[i] = 0.f;
}

// rm[n] = mean_s x[n,s]   (coalesced streaming pass over 105 MB)
__global__ void k_mean(const float* __restrict__ x, float* __restrict__ ws) {
  const int n = blockIdx.x, t = threadIdx.x;
  const float* p = x + (size_t)n * S_TOT * 256 + t;
  float s = 0.f;
  for (int i = 0; i < S_TOT; ++i) s += p[(size_t)i * 256];
  ws[OFF_RM + n * 256 + t] = s * (1.f / (float)S_TOT);
}

// blocks 0..31: sq1/isq1 = sqrtm/invsqrtm(rm[n]); block 32: Wsq=expm(0.25(W+Wt))
__global__ __launch_bounds__(32) void k_prep1(const float* __restrict__ W,
                                              float* __restrict__ ws) {
  __shared__ __align__(16) float sm[6 * 256];
  const int lane = threadIdx.x & 31;
  float* s0 = sm;
  float* s1 = sm + 256;
  float* s2 = sm + 512;
  float* s3 = sm + 768;
  if (blockIdx.x < N_B) {
    const int n = blockIdx.x;
    for (int i = 0; i < 8; ++i) {
      int e = lane * 8 + i;
      s0[e] = ws[OFF_RM + n * 256 + e];
    }
    spd_sqrt_invsqrt(s0, s1, s2, s3, lane);
    for (int i = 0; i < 8; ++i) {
      int e = lane * 8 + i;
      ws[OFF_SQ1 + n * 256 + e] = s1[e];
      ws[OFF_ISQ1 + n * 256 + e] = s2[e];
    }
  } else {
    for (int i = 0; i < 8; ++i) {
      int e = lane * 8 + i, r = e >> 4, cc = e & 15;
      s0[e] = 0.25f * (W[r * 16 + cc] + W[cc * 16 + r]);  // 0.5*sym(W)
    }
    spd_expm(s0, s1, s2, s3, lane, 1.f);  // Wsq = expm(0.5 sym(W))
    for (int i = 0; i < 8; ++i) {
      int e = lane * 8 + i;
      ws[OFF_WSQ + e] = s1[e];
    }
  }
}

// gt[n] += logm(isq1 X isq1) summed over samples (4 samples per block)
__global__ __launch_bounds__(128) void k_gt(const float* __restrict__ x,
                                            float* __restrict__ ws) {
  __shared__ __align__(16) float sm[4 * 6 * 256];
  const int lane = threadIdx.x & 31, w = threadIdx.x >> 5;
  const int n = blockIdx.x;
  const int s = blockIdx.y * 4 + w;
  float* s0 = sm + w * 1536;        // X -> B
  float* s1 = s0 + 256;             // isq1
  float* sL = s0 + 512;             // logm result
  float* t1 = s0 + 768;
  float* t2 = s0 + 1024;
  float* t3 = s0 + 1280;
  const float* xp = x + ((size_t)n * S_TOT + s) * 256;
  async_load_1k(xp, s0, lane);      // X -> LDS, no VGPR round-trip
  for (int i = 0; i < 8; ++i) {     // overlaps with the async copy
    int e = lane * 8 + i;
    s1[e] = ws[OFF_ISQ1 + n * 256 + e];
  }
  wait_async0();
  mm16(s1, s0, t1, lane);
  mm16(t1, s1, s0, lane);           // B = isq1 X isq1
  spd_logm(s0, sL, t1, t2, t3, lane);
  __syncthreads();
  if (w == 0) {
    float* gt = ws + OFF_GT + n * 256;
    for (int i = 0; i < 8; ++i) {
      int e = lane * 8 + i;
      float v = sm[0 * 1536 + 512 + e] + sm[1 * 1536 + 512 + e] +
                sm[2 * 1536 + 512 + e] + sm[3 * 1536 + 512 + e];
      atomicAdd(&gt[e], v);
    }
  }
}

// x_mean = sq1 expm(gt/S) sq1 ; sq2/isq2 = sqrt/invsqrt(x_mean);
// ltmean = lower_triangle(logm(x_mean))
__global__ __launch_bounds__(32) void k_finish_mean(float* __restrict__ ws) {
  __shared__ __align__(16) float sm[8 * 256];
  const int lane = threadIdx.x & 31, n = blockIdx.x;
  float* s0 = sm;
  float* s1 = sm + 256;
  float* s2 = sm + 512;
  float* s3 = sm + 768;
  float* s4 = sm + 1024;
  float* s5 = sm + 1280;
  float* s6 = sm + 1536;
  for (int i = 0; i < 8; ++i) {
    int e = lane * 8 + i;
    s0[e] = ws[OFF_GT + n * 256 + e] * (1.f / (float)S_TOT);
  }
  spd_expm(s0, s1, s2, s3, lane, 1.f);       // s1 = expm(mean log)
  for (int i = 0; i < 8; ++i) {
    int e = lane * 8 + i;
    s2[e] = ws[OFF_SQ1 + n * 256 + e];
  }
  mm16(s2, s1, s4, lane);
  mm16(s4, s2, s1, lane);                    // s1 = x_mean
  for (int i = 0; i < 8; ++i) {
    int e = lane * 8 + i;
    ws[OFF_XMEAN + n * 256 + e] = s1[e];
  }
  spd_sqrt_invsqrt(s1, s4, s5, s6, lane);    // s4=sq2, s5=isq2
  for (int i = 0; i < 8; ++i) {
    int e = lane * 8 + i;
    ws[OFF_SQ2 + n * 256 + e] = s4[e];
    ws[OFF_ISQ2 + n * 256 + e] = s5[e];
    s0[e] = s1[e];                           // copy x_mean for logm
  }
  spd_logm(s0, s2, s3, s4, s5, lane);        // s2 = logm(x_mean)
  for (int idx = lane; idx < D0; idx += 32) {
    int r, c;
    tri_rc(idx, &r, &c);
    float v = (r == c) ? s2[r * 17] : SQRT2 * s2[r * 16 + c];
    ws[OFF_LTMEAN + n * D0 + idx] = v;
  }
}

// xc[n,s] = lower_triangle(Wsq logm(isq2 X isq2) Wsq), zero-padded to 160
__global__ __launch_bounds__(128) void k_center(const float* __restrict__ x,
                                                float* __restrict__ ws) {
  __shared__ __align__(16) float sm[4 * 7 * 256];
  const int lane = threadIdx.x & 31, w = threadIdx.x >> 5;
  const int n = blockIdx.x;
  const int s = blockIdx.y * 4 + w;
  float* s0 = sm + w * 1792;        // X -> B
  float* s1 = s0 + 256;             // isq2
  float* s2 = s0 + 512;             // Wsq
  float* sL = s0 + 768;             // result
  float* t1 = s0 + 1024;
  float* t2 = s0 + 1280;
  float* t3 = s0 + 1536;
  const float* xp = x + ((size_t)n * S_TOT + s) * 256;
  async_load_1k(xp, s0, lane);      // X -> LDS, no VGPR round-trip
  for (int i = 0; i < 8; ++i) {     // overlaps with the async copy
    int e = lane * 8 + i;
    s1[e] = ws[OFF_ISQ2 + n * 256 + e];
    s2[e] = ws[OFF_WSQ + e];
  }
  wait_async0();
  mm16(s1, s0, t1, lane);
  mm16(t1, s1, s0, lane);           // B = isq2 X isq2
  spd_logm(s0, sL, t1, t2, t3, lane);
  mm16(s2, sL, t1, lane);
  mm16(t1, s2, sL, lane);           // x_center = Wsq L Wsq
  float* xcrow = ws + OFF_XC + ((size_t)n * S_TOT + s) * DP;
  for (int idx = lane; idx < DP; idx += 32) {
    float v = 0.f;
    if (idx < D0) {
      int r, c;
      tri_rc(idx, &r, &c);
      v = (r == c) ? sL[r * 17] : SQRT2 * sL[r * 16 + c];
    }
    xcrow[idx] = v;
  }
}

// cov[n] = xc[n]^T xc[n] / (S-1), one 16x16 tile per wave, K=3200
__global__ __launch_bounds__(128) void k_cov(float* __restrict__ ws) {
  const int lane = threadIdx.x & 31, w = threadIdx.x >> 5;
  const int n = blockIdx.x;
  const int t = blockIdx.y * 4 + w;
  const int ti = t / 10, tj = t % 10;
  const int m = lane & 15, h = lane >> 4;
  const float* xc = ws + OFF_XC + (size_t)n * S_TOT * DP;
  v8f acc = {};
  for (int k = 0; k < S_TOT; k += 4) {
    const float* r0 = xc + (size_t)(k + 2 * h) * DP;
    const float* r1 = r0 + DP;
    v2f a, b;
    a.x = r0[ti * 16 + m];
    a.y = r1[ti * 16 + m];
    b.x = r0[tj * 16 + m];
    b.y = r1[tj * 16 + m];
    acc = __builtin_amdgcn_wmma_f32_16x16x4_f32(false, a, false, b, (short)0,
                                                acc, false, false);
  }
  float* cov = ws + OFF_COV + (size_t)n * DP * DP;
  const float sc = 1.f / (float)(S_TOT - 1);
#pragma unroll
  for (int r = 0; r < 8; ++r)
    cov[(ti * 16 + r + 8 * h) * DP + tj * 16 + m] = acc[r] * sc;
}

// trace-normalize + 1e-5 I + in-place Cholesky (136x136 block of cov[n])
__global__ __launch_bounds__(256) void k_chol_cov(float* __restrict__ ws) {
  __shared__ float red[256];
  const int tid = threadIdx.x, n = blockIdx.x;
  float* A = ws + OFF_COV + (size_t)n * DP * DP;
  float ts = 0.f;
  for (int i = tid; i < D0; i += 256) ts += A[i * DP + i];
  red[tid] = ts;
  __syncthreads();
  for (int st = 128; st > 0; st >>= 1) {
    if (tid < st) red[tid] += red[tid + st];
    __syncthreads();
  }
  const float inv_tr = 1.f / red[0];
  __syncthreads();
  for (int e = tid; e < D0 * D0; e += 256) {
    int r = e / D0, c = e % D0;
    A[r * DP + c] = A[r * DP + c] * inv_tr + ((r == c) ? 1e-5f : 0.f);
  }
  __syncthreads();
  for (int j = 0; j < D0; ++j) {
    if (tid == 0) A[j * DP + j] = sqrtf(fmaxf(A[j * DP + j], 1e-20f));
    __syncthreads();
    const float dj = A[j * DP + j];
    for (int i = j + 1 + tid; i < D0; i += 256) A[i * DP + j] /= dj;
    __syncthreads();
    for (int i = j + 1 + tid; i < D0; i += 256) {
      const float lij = A[i * DP + j];
      for (int c2 = j + 1; c2 <= i; ++c2) A[i * DP + c2] -= lij * A[c2 * DP + j];
    }
    __syncthreads();
  }
}

// 160x160 block matmul on global buffers, 8 waves, WMMA-tiled
__device__ void block_mm160(const float* A, const float* B, float* D, int lane,
                            int w) {
  const int m = lane & 15, h = lane >> 4;
  for (int t = w; t < 100; t += 8) {
    const int ti = t / 10, tj = t % 10;
    v8f acc = {};
    for (int k = 0; k < DP; k += 4) {
      v2f a, b;
      a.x = A[(ti * 16 + m) * DP + k + 2 * h];
      a.y = A[(ti * 16 + m) * DP + k + 2 * h + 1];
      b.x = B[(k + 2 * h) * DP + tj * 16 + m];
      b.y = B[(k + 2 * h + 1) * DP + tj * 16 + m];
      acc = __builtin_amdgcn_wmma_f32_16x16x4_f32(false, a, false, b, (short)0,
                                                  acc, false, false);
    }
#pragma unroll
    for (int r = 0; r < 8; ++r)
      D[(ti * 16 + r + 8 * h) * DP + tj * 16 + m] = acc[r];
  }
}

// R = chol(expm(sym(W_lw))) padded to 160 (pad block = I), upper zeroed
__global__ __launch_bounds__(256) void k_rlw(const float* __restrict__ Wlw,
                                             float* __restrict__ ws) {
  const int tid = threadIdx.x, lane = tid & 31, w = tid >> 5;
  float* S = ws + OFF_RBIG;
  float* A = ws + OFF_SCR0;
  float* P = ws + OFF_SCR1;
  float* T = ws + OFF_SCR2;
  for (int e = tid; e < DP * DP; e += 256) {
    int r = e / DP, c = e % DP;
    float v = 0.f;
    if (r < D_FULL && c < D_FULL)
      v = 0.5f * (Wlw[r * D_FULL + c] + Wlw[c * D_FULL + r]) *
          (1.f / (float)(1 << EXP_SQ));
    A[e] = v;
  }
  __syncthreads();
  for (int e = tid; e < DP * DP; e += 256) {
    int r = e / DP, c = e % DP;
    S[e] = ((r == c) ? 1.f : 0.f) + A[e];
    P[e] = A[e];
  }
  __syncthreads();
  for (int k = 2; k <= EXP_TERMS; ++k) {
    block_mm160(P, A, T, lane, w);
    __syncthreads();
    const float ik = 1.f / (float)k;
    for (int e = tid; e < DP * DP; e += 256) {
      float pv = T[e] * ik;
      P[e] = pv;
      S[e] += pv;
    }
    __syncthreads();
  }
  block_mm160(S, S, T, lane, w);
  __syncthreads();
  block_mm160(T, T, S, lane, w);
  __syncthreads();
  block_mm160(S, S, T, lane, w);
  __syncthreads();
  block_mm160(T, T, S, lane, w);
  __syncthreads();
  for (int j = 0; j < DP; ++j) {
    if (tid == 0) S[j * DP + j] = sqrtf(fmaxf(S[j * DP + j], 1e-20f));
    __syncthreads();
    const float dj = S[j * DP + j];
    for (int i = j + 1 + tid; i < DP; i += 256) S[i * DP + j] /= dj;
    __syncthreads();
    for (int i = j + 1 + tid; i < DP; i += 256) {
      const float lij = S[i * DP + j];
      for (int c2 = j + 1; c2 <= i; ++c2) S[i * DP + c2] -= lij * S[c2 * DP + j];
    }
    __syncthreads();
  }
  for (int e = tid; e < DP * DP; e += 256) {
    int r = e / DP, c = e % DP;
    if (c > r) S[e] = 0.f;
  }
}

// E[n]: [[L,0],[ltmean (x16 rows), I_k]], padded to 160 (all entries written)
__global__ __launch_bounds__(256) void k_assembleE(float* __restrict__ ws) {
  const int n = blockIdx.x, tid = threadIdx.x;
  const float* L = ws + OFF_COV + (size_t)n * DP * DP;
  const float* lt = ws + OFF_LTMEAN + n * D0;
  float* E = ws + OFF_E + (size_t)n * DP * DP;
  for (int e = tid; e < DP * DP; e += 256) {
    int r = e / DP, c = e % DP;
    float v = 0.f;
    if (r < D0 && c < D0)
      v = (c <= r) ? L[r * DP + c] : 0.f;
    else if (r >= D0 && r < D_FULL && c < D0)
      v = lt[c];
    else if (r >= D0 && r < D_FULL && r == c)
      v = 1.f;
    E[e] = v;
  }
}

// out[n] = (E[n] @ R)[0:152, 0:152]
__global__ __launch_bounds__(128) void k_final(float* __restrict__ ws,
                                               float* __restrict__ out) {
  const int lane = threadIdx.x & 31, w = threadIdx.x >> 5;
  const int n = blockIdx.x;
  const int t = blockIdx.y * 4 + w;
  const int ti = t / 10, tj = t % 10;
  const int m = lane & 15, h = lane >> 4;
  const float* E = ws + OFF_E + (size_t)n * DP * DP;
  const float* R = ws + OFF_RBIG;
  v8f acc = {};
  for (int k = 0; k < DP; k += 4) {
    v2f a, b;
    a.x = E[(ti * 16 + m) * DP + k + 2 * h];
    a.y = E[(ti * 16 + m) * DP + k + 2 * h + 1];
    b.x = R[(k + 2 * h) * DP + tj * 16 + m];
    b.y = R[(k + 2 * h + 1) * DP + tj * 16 + m];
    acc = __builtin_amdgcn_wmma_f32_16x16x4_f32(false, a, false, b, (short)0,
                                                acc, false, false);
  }
#pragma unroll
  for (int r = 0; r < 8; ++r) {
    const int row = ti * 16 + r + 8 * h, col = tj * 16 + m;
    if (row < D_FULL && col < D_FULL)
      out[(size_t)n * D_FULL * D_FULL + row * D_FULL + col] = acc[r];
  }
}

// ---------------------------------------------------------------------------
extern "C" void kernel_launch(void* const* d_in, const int* in_sizes, int n_in,
                              void* d_out, int out_size, void* d_ws,
                              size_t ws_size, hipStream_t stream) {
  (void)in_sizes; (void)n_in; (void)out_size; (void)ws_size;
  const float* x   = (const float*)d_in[0];
  const float* W   = (const float*)d_in[1];
  const float* Wlw = (const float*)d_in[2];
  float* out = (float*)d_out;
  float* ws  = (float*)d_ws;

  k_zero<<<32, 256, 0, stream>>>(ws + OFF_GT, 32 * 256);
  k_mean<<<N_B, 256, 0, stream>>>(x, ws);
  k_prep1<<<N_B + 1, 32, 0, stream>>>(W, ws);
  k_gt<<<dim3(N_B, S_TOT / 4), 128, 0, stream>>>(x, ws);
  k_finish_mean<<<N_B, 32, 0, stream>>>(ws);
  k_center<<<dim3(N_B, S_TOT / 4), 128, 0, stream>>>(x, ws);
  k_rlw<<<1, 256, 0, stream>>>(Wlw, ws);
  k_cov<<<dim3(N_B, 25), 128, 0, stream>>>(ws);
  k_chol_cov<<<N_B, 256, 0, stream>>>(ws);
  k_assembleE<<<N_B, 256, 0, stream>>>(ws);
  k_final<<<dim3(N_B, 25), 128, 0, stream>>>(ws, out);
}